// RecursiveNN_50723563765863
// MI455X (gfx1250) — compile-verified
//
#include <hip/hip_runtime.h>
#include <stdint.h>

// Problem constants (match reference).
#define Bn      1024
#define Vn      1024
#define Ln      16
#define Wn      8192
#define NLEAF   2048                      // 2*V
#define NTOTAL  (NLEAF + Ln * Wn)         // 133120 nodes
#define BM_DW   (NTOTAL / 32)             // 4160 dwords of reachability bitmap
#define VALS_OFF 65536                    // vals table starts 64KB into d_ws
// d_ws layout: [0,16640): bitmap | [64KB, 64KB + NTOTAL*Bn*4): node-major value table.
// Only columns of *needed* nodes are ever touched (DCE), so physical traffic is tiny.

// ---- CDNA5 async global->LDS path (guarded; falls back to direct loads) ----
#if __has_builtin(__builtin_amdgcn_global_load_async_to_lds_b128)
#define HAVE_ASYNC_B128 1
#else
#define HAVE_ASYNC_B128 0
#endif

#if __has_builtin(__builtin_amdgcn_s_wait_asynccnt)
#define WAIT_ASYNC0() __builtin_amdgcn_s_wait_asynccnt(0)
#else
#define WAIT_ASYNC0() asm volatile("s_wait_asynccnt 0" ::: "memory")
#endif

#if HAVE_ASYNC_B128
typedef int v4i_t __attribute__((ext_vector_type(4)));
typedef __attribute__((address_space(1))) v4i_t g_v4i;   // global int4
typedef __attribute__((address_space(3))) v4i_t l_v4i;   // LDS int4

__device__ __forceinline__ void async_cp16(const float* src, const float* lds_dst) {
  // (global src, lds dst, imm offset, imm cpol). Low 32 bits of the flat LDS
  // address are the LDS byte offset (aperture form), so the inttoptr is exact.
  __builtin_amdgcn_global_load_async_to_lds_b128(
      (g_v4i*)(uintptr_t)src,
      (l_v4i*)(uint32_t)(uintptr_t)lds_dst,
      0, 0);
}
#endif

// ---------------------------------------------------------------------------
// Backward reachability: mark transitive fan-in of the root node (TOTAL-1).
// Single workgroup; bitmap lives in LDS (16.6 KB), ds_or_b32 atomics.
// Layer l's own bits are only written by layers > l, so one barrier per layer
// suffices (reads at [base, base+W) never alias writes at [0, base)).
// ---------------------------------------------------------------------------
__global__ void __launch_bounds__(1024)
mark_kernel(const int* __restrict__ li, const int* __restrict__ ri,
            unsigned* __restrict__ bm_out) {
  __shared__ unsigned mb[BM_DW];
  const int t = threadIdx.x;
  for (int i = t; i < BM_DW; i += 1024) mb[i] = 0u;
  __syncthreads();
  if (t == 0) mb[(NTOTAL - 1) >> 5] = 1u << ((NTOTAL - 1) & 31);
  __syncthreads();
  for (int l = Ln - 1; l >= 0; --l) {
    const int base = NLEAF + l * Wn;   // == avail for layer l
    for (int g = t; g < Wn; g += 1024) {
      const int node = base + g;
      if ((mb[node >> 5] >> (node & 31)) & 1u) {
        const unsigned a = (unsigned)li[l * Wn + g] % (unsigned)base;
        const unsigned c = (unsigned)ri[l * Wn + g] % (unsigned)base;
        atomicOr(&mb[a >> 5], 1u << (a & 31));
        atomicOr(&mb[c >> 5], 1u << (c & 31));
      }
    }
    __syncthreads();
  }
  for (int i = t; i < BM_DW; i += 1024) bm_out[i] = mb[i];
}

// ---------------------------------------------------------------------------
// Materialize only the needed leaf columns, node-major ([node][batch]) so every
// downstream gather is one contiguous 4 KB block.
// ---------------------------------------------------------------------------
__global__ void __launch_bounds__(256)
leaf_kernel(const float* __restrict__ x, const unsigned* __restrict__ bm,
            float* __restrict__ vals) {
  const int n = blockIdx.x;            // leaf node id, 0..2047
  if (!((bm[n >> 5] >> (n & 31)) & 1u)) return;
  const int t = threadIdx.x;           // 0..255
  const int col = (n < Vn) ? n : (n - Vn);
  float* dst = vals + (size_t)n * Bn;
  #pragma unroll
  for (int k = 0; k < 4; ++k) {
    const int b = t + k * 256;
    const float v = x[(size_t)b * Vn + col];
    dst[b] = (n < Vn) ? v : 1.0f - v;  // leaf n>=V is 1-x
  }
}

// ---------------------------------------------------------------------------
// One layer of gates, 8 gates per block, fully predicated on the bitmap.
// Needed gates stage both 4 KB source columns into LDS via the async pipe
// (each wave consumes exactly the slice it loaded -> no barrier for staging).
// ---------------------------------------------------------------------------
__global__ void __launch_bounds__(256)
gate_kernel(const int* __restrict__ li, const int* __restrict__ ri,
            const int* __restrict__ opv, const unsigned* __restrict__ bm,
            float* __restrict__ vals, int avail) {
  __shared__ float sm[2 * Bn];         // left column | right column (8 KB)
  const int t  = threadIdx.x;          // 0..255, 4 floats each
  const int g0 = blockIdx.x * 8;
  #pragma unroll 1
  for (int i = 0; i < 8; ++i) {
    const int g    = g0 + i;
    const int node = avail + g;
    // Uniform predicate across the block (same node), so barriers stay legal.
    if (!((bm[node >> 5] >> (node & 31)) & 1u)) continue;
    const unsigned a = (unsigned)li[g] % (unsigned)avail;
    const unsigned c = (unsigned)ri[g] % (unsigned)avail;
    const float* pa = vals + (size_t)a * Bn + t * 4;
    const float* pc = vals + (size_t)c * Bn + t * 4;
    float4 va, vc;
#if HAVE_ASYNC_B128
    async_cp16(pa, &sm[t * 4]);
    async_cp16(pc, &sm[Bn + t * 4]);
    WAIT_ASYNC0();
    va = *(const float4*)&sm[t * 4];
    vc = *(const float4*)&sm[Bn + t * 4];
#else
    va = *(const float4*)pa;
    vc = *(const float4*)pc;
#endif
    float4 r;
    if (opv[g] == 1) {
      r.x = va.x + vc.x; r.y = va.y + vc.y; r.z = va.z + vc.z; r.w = va.w + vc.w;
    } else {
      r.x = va.x * vc.x; r.y = va.y * vc.y; r.z = va.z * vc.z; r.w = va.w * vc.w;
    }
    *(float4*)(vals + (size_t)node * Bn + t * 4) = r;
#if HAVE_ASYNC_B128
    __syncthreads();                   // don't overwrite sm before all reads retire
#endif
  }
}

// Root column -> d_out (reference returns vals[:, -1], float32 [B]).
__global__ void __launch_bounds__(256)
out_kernel(const float* __restrict__ vals, float* __restrict__ out) {
  const int t = threadIdx.x;
  *(float4*)(out + t * 4) =
      *(const float4*)(vals + (size_t)(NTOTAL - 1) * Bn + t * 4);
}

extern "C" void kernel_launch(void* const* d_in, const int* in_sizes, int n_in,
                              void* d_out, int out_size, void* d_ws, size_t ws_size,
                              hipStream_t stream) {
  (void)in_sizes; (void)n_in; (void)out_size; (void)ws_size;
  const float* x  = (const float*)d_in[0];
  const int*   li = (const int*)d_in[1];
  const int*   ri = (const int*)d_in[2];
  const int*   op = (const int*)d_in[3];
  unsigned* bm   = (unsigned*)d_ws;
  float*    vals = (float*)((char*)d_ws + VALS_OFF);
  float*    out  = (float*)d_out;

  // 1) DCE: backward reachability from the root.
  mark_kernel<<<1, 1024, 0, stream>>>(li, ri, bm);
  // 2) Needed leaves only, transposed to node-major.
  leaf_kernel<<<NLEAF, 256, 0, stream>>>(x, bm, vals);
  // 3) 16 dependent layers; each launch computes only marked gates.
  for (int l = 0; l < Ln; ++l) {
    const int avail = NLEAF + l * Wn;
    gate_kernel<<<Wn / 8, 256, 0, stream>>>(li + l * Wn, ri + l * Wn,
                                            op + l * Wn, bm, vals, avail);
  }
  // 4) Emit root column.
  out_kernel<<<1, 256, 0, stream>>>(vals, out);
}